// MemoryModule_3667902071294
// MI455X (gfx1250) — compile-verified
//
#include <hip/hip_runtime.h>
#include <hip/hip_bf16.h>

typedef float v2f __attribute__((ext_vector_type(2)));
typedef float v4f __attribute__((ext_vector_type(4)));
typedef float v8f __attribute__((ext_vector_type(8)));

#define NUM_NODES 500000
#define DIM       172
#define BATCH     4096
#define NPAD      176            // per-gate N padded to multiple of 16
#define WROW      (3 * NPAD)     // 528 columns per k-pair row
#define KPAIRS    (DIM / 2)      // 86 k-pairs (DIM is even)
#define M_TILE    32             // batch rows per workgroup (2 waves x 16)

// ---------------------------------------------------------------------------
// Bulk streaming copy: memory -> out_mem, last_update -> out_lu.
// Bandwidth-dominant part (~690 MB); b128 nontemporal ops.
// ---------------------------------------------------------------------------
__global__ void tgn_copy(const v4f* __restrict__ src_mem,
                         const v4f* __restrict__ src_lu,
                         v4f* __restrict__ dst_mem,
                         v4f* __restrict__ dst_lu) {
  const long long MEM4 = (long long)NUM_NODES * (DIM / 4);  // 21,500,000
  const long long LU4  = NUM_NODES / 4;                     //    125,000
  const long long stride = (long long)gridDim.x * blockDim.x;
  for (long long i = (long long)blockIdx.x * blockDim.x + threadIdx.x;
       i < MEM4 + LU4; i += stride) {
    if (i < MEM4) {
      v4f v = __builtin_nontemporal_load(&src_mem[i]);
      __builtin_nontemporal_store(v, &dst_mem[i]);
    } else {
      v4f v = __builtin_nontemporal_load(&src_lu[i - MEM4]);
      __builtin_nontemporal_store(v, &dst_lu[i - MEM4]);
    }
  }
}

// ---------------------------------------------------------------------------
// One-time weight transpose into workspace, k-PAIR interleaved:
//   wt2[mat][k/2][gate*NPAD + col][j] = w[mat][(gate*DIM + col)][2*(k/2)+j]
// so each lane's WMMA B fragment {K=kb, K=kb+1} is one contiguous b64 load,
// and lanes 0-15 (consecutive col) cover a contiguous 128B segment.
// ---------------------------------------------------------------------------
__global__ void tgn_prep_wt(const float* __restrict__ w_ih,
                            const float* __restrict__ w_hh,
                            float* __restrict__ wt) {
  const int idx = blockIdx.x * blockDim.x + threadIdx.x;
  const int total = 2 * KPAIRS * WROW * 2;
  if (idx >= total) return;
  const int j    = idx & 1;
  const int col  = (idx >> 1) % WROW;
  const int pair = ((idx >> 1) / WROW) % KPAIRS;
  const int mat  = idx / (2 * WROW * KPAIRS);
  const int g    = col / NPAD;
  const int cc   = col - g * NPAD;
  const int k    = 2 * pair + j;
  const float* w = mat ? w_hh : w_ih;
  wt[idx] = (cc < DIM) ? w[(g * DIM + cc) * DIM + k] : 0.0f;
}

// ---------------------------------------------------------------------------
// Fused gather + dual GEMM (fp32 WMMA 16x16x4) + GRU gates + scatter.
// 64 threads = 2 waves; each wave owns 16 batch rows. 6 v8f accumulators
// (i_r,i_z,i_n,h_r,h_z,h_n) stay in registers; gi/gh never hit memory.
// ---------------------------------------------------------------------------
__launch_bounds__(64)
__global__ void tgn_gru(const int*   __restrict__ node_ids,
                        const float* __restrict__ messages,
                        const float* __restrict__ timestamps,
                        const float* __restrict__ memory,
                        const float* __restrict__ b_ih,
                        const float* __restrict__ b_hh,
                        const float* __restrict__ wt,   // [2][KPAIRS][WROW][2]
                        float* __restrict__ out_mem,
                        float* __restrict__ out_lu) {
  __shared__ float Xs[M_TILE * DIM];   // 22 KB
  __shared__ float Hs[M_TILE * DIM];   // 22 KB
  __shared__ int   nid[M_TILE];
  __shared__ int   win[M_TILE];

  const int tid  = threadIdx.x;
  const int b0   = blockIdx.x * M_TILE;
  const int lane = tid & 31;
  const int wave = tid >> 5;     // 0..1
  const int hi   = lane >> 4;    // lane half (K/M split in WMMA layouts)
  const int lq   = lane & 15;

  // Phase 0: ids, global last-occurrence winner flags, last_update scatter.
  if (tid < M_TILE) {
    const int b  = b0 + tid;
    const int id = node_ids[b];
    nid[tid] = id;
    int w = 1;
    for (int bb = b + 1; bb < BATCH; ++bb)
      if (node_ids[bb] == id) { w = 0; break; }
    win[tid] = w;
    if (w) out_lu[id] = timestamps[b];
  }
  // Stage X tile (contiguous rows).
  for (int i = tid; i < M_TILE * DIM; i += 64)
    Xs[i] = messages[(long long)b0 * DIM + i];
  __syncthreads();
  // Stage gathered H tile.
  for (int i = tid; i < M_TILE * DIM; i += 64) {
    const int r = i / DIM, c = i - r * DIM;
    Hs[i] = memory[(long long)nid[r] * DIM + c];
  }
  __syncthreads();

  const int rowA = wave * 16 + lq;          // A-fragment source row (M = lq)
  const float* __restrict__ wtih = wt;
  const float* __restrict__ wthh = wt + KPAIRS * WROW * 2;

  for (int tc = 0; tc < 11; ++tc) {         // 11 * 16 = 176 >= 172 cols
    const int  c    = tc * 16;
    const int  col  = c + lq;               // this lane's output column
    const bool colv = col < DIM;

    float bir = 0.f, biz = 0.f, bin = 0.f, bhr = 0.f, bhz = 0.f, bhn = 0.f;
    if (colv) {
      bir = b_ih[col]; biz = b_ih[DIM + col]; bin = b_ih[2 * DIM + col];
      bhr = b_hh[col]; bhz = b_hh[DIM + col]; bhn = b_hh[2 * DIM + col];
    }

    v8f air = {}, aiz = {}, ain = {}, ahr = {}, ahz = {}, ahn = {};

    for (int k0 = 0; k0 < DIM; k0 += 4) {   // 43 exact K-steps
      const int kb = k0 + 2 * hi;           // A/B vgpr0 -> K=kb, vgpr1 -> K=kb+1
      const v2f ax = *(const v2f*)&Xs[rowA * DIM + kb];   // 8B LDS pair
      const v2f ah = *(const v2f*)&Hs[rowA * DIM + kb];
      // k-pair interleaved weights: one b64 per B fragment, fully coalesced.
      const int w0 = (kb >> 1) * (WROW * 2) + 2 * col;
      const v2f bir2 = *(const v2f*)&wtih[w0];
      const v2f biz2 = *(const v2f*)&wtih[w0 + 2 * NPAD];
      const v2f bin2 = *(const v2f*)&wtih[w0 + 4 * NPAD];
      const v2f bhr2 = *(const v2f*)&wthh[w0];
      const v2f bhz2 = *(const v2f*)&wthh[w0 + 2 * NPAD];
      const v2f bhn2 = *(const v2f*)&wthh[w0 + 4 * NPAD];
      air = __builtin_amdgcn_wmma_f32_16x16x4_f32(false, ax, false, bir2, (short)0, air, false, false);
      aiz = __builtin_amdgcn_wmma_f32_16x16x4_f32(false, ax, false, biz2, (short)0, aiz, false, false);
      ain = __builtin_amdgcn_wmma_f32_16x16x4_f32(false, ax, false, bin2, (short)0, ain, false, false);
      ahr = __builtin_amdgcn_wmma_f32_16x16x4_f32(false, ah, false, bhr2, (short)0, ahr, false, false);
      ahz = __builtin_amdgcn_wmma_f32_16x16x4_f32(false, ah, false, bhz2, (short)0, ahz, false, false);
      ahn = __builtin_amdgcn_wmma_f32_16x16x4_f32(false, ah, false, bhn2, (short)0, ahn, false, false);
    }

    // GRU gates directly on accumulator registers; scatter winning rows.
    // C/D layout: vgpr e, lanes 0-15 -> M=e, lanes 16-31 -> M=e+8; N=lq.
    #pragma unroll
    for (int e = 0; e < 8; ++e) {
      const int rt = wave * 16 + e + 8 * hi;   // row within tile
      const float hprev = colv ? Hs[rt * DIM + col] : 0.0f;
      const float xr = air[e] + bir + ahr[e] + bhr;
      const float xz = aiz[e] + biz + ahz[e] + bhz;
      const float r  = 1.0f / (1.0f + expf(-xr));
      const float z  = 1.0f / (1.0f + expf(-xz));
      const float n  = tanhf(ain[e] + bin + r * (ahn[e] + bhn));
      const float hn = (1.0f - z) * n + z * hprev;
      if (colv && win[rt])
        out_mem[(long long)nid[rt] * DIM + col] = hn;
    }
  }
}

// ---------------------------------------------------------------------------
extern "C" void kernel_launch(void* const* d_in, const int* in_sizes, int n_in,
                              void* d_out, int out_size, void* d_ws, size_t ws_size,
                              hipStream_t stream) {
  const int*   node_ids    = (const int*)  d_in[0];
  const float* messages    = (const float*)d_in[1];
  const float* timestamps  = (const float*)d_in[2];
  const float* memory      = (const float*)d_in[3];
  const float* last_update = (const float*)d_in[4];
  const float* w_ih        = (const float*)d_in[5];
  const float* w_hh        = (const float*)d_in[6];
  const float* b_ih        = (const float*)d_in[7];
  const float* b_hh        = (const float*)d_in[8];

  float* out_mem = (float*)d_out;
  float* out_lu  = out_mem + (long long)NUM_NODES * DIM;   // 16B-aligned offset
  float* wt      = (float*)d_ws;   // needs 2*KPAIRS*WROW*2*4 = 726,528 bytes

  // 1) transpose+pad+pair-interleave weights into workspace (tiny, one pass)
  tgn_prep_wt<<<(2 * KPAIRS * WROW * 2 + 255) / 256, 256, 0, stream>>>(w_ih, w_hh, wt);
  // 2) bandwidth-dominant full-table copy (memory + last_update -> d_out)
  tgn_copy<<<2048, 256, 0, stream>>>((const v4f*)memory, (const v4f*)last_update,
                                     (v4f*)out_mem, (v4f*)out_lu);
  // 3) fused gather + WMMA GRU + scatter (overwrites the 4096 touched rows)
  tgn_gru<<<BATCH / M_TILE, 64, 0, stream>>>(node_ids, messages, timestamps,
                                             memory, b_ih, b_hh, wt,
                                             out_mem, out_lu);
}